// FSTHeadwiseEntmaxScorer_15805479649712
// MI455X (gfx1250) — compile-verified
//
#include <hip/hip_runtime.h>

typedef __attribute__((ext_vector_type(2))) float v2f;
typedef __attribute__((ext_vector_type(8))) float v8f;

#define Bb 8
#define Nn 256
#define Dd 128
#define Hh 128
#define Kk 8

// ---------------------------------------------------------------------------
// Kernel 1: HI[k][b][n][h] = E[b][n][:] . W1[k][0:D][h]
//           HJ[k][b][n][h] = E[b][n][:] . W1[k][D:2D][h] + b1[k][h]
// One wave (32 threads) per 16x16 output tile, V_WMMA_F32_16X16X4_F32,
// 32 accumulation steps of K=4 over D=128.
// ---------------------------------------------------------------------------
__global__ __launch_bounds__(32)
void fst_gemm_wmma(const float* __restrict__ E, const float* __restrict__ W1,
                   const float* __restrict__ b1,
                   float* __restrict__ HI, float* __restrict__ HJ)
{
    const int nt = Hh / 16;   // 8 h-tiles
    const int mt = Nn / 16;   // 16 n-tiles
    int blk = blockIdx.x;
    int ht = blk % nt; blk /= nt;
    int it = blk % mt; blk /= mt;
    int ij = blk & 1;  blk >>= 1;
    int b  = blk % Bb; blk /= Bb;
    int k  = blk;             // head 0..7

    const int lane = threadIdx.x;      // 0..31
    const int m    = lane & 15;        // M row (A) / N col (B)
    const int kq   = (lane >> 4) * 2;  // 0 or 2: which K pair this half-wave holds

    const float* Ep = E + ((size_t)b * Nn + (size_t)it * 16 + m) * Dd;
    const float* Wbase = W1 + ((size_t)k * 2 * Dd + (size_t)ij * Dd) * Hh + (size_t)ht * 16 + m;

    v8f c = {};
    #pragma unroll 4
    for (int d0 = 0; d0 < Dd; d0 += 4) {
        v2f av, bv;
        // A 16x4 f32: lane m holds M=m; VGPR0=K(d0+kq), VGPR1=K(d0+kq+1)
        av.x = Ep[d0 + kq];
        av.y = Ep[d0 + kq + 1];
        // B 4x16 f32: lane m holds N=m; VGPR0=K(d0+kq) row, VGPR1=K(d0+kq+1) row
        const float* wb = Wbase + (size_t)(d0 + kq) * Hh;
        bv.x = wb[0];
        bv.y = wb[Hh];
        c = __builtin_amdgcn_wmma_f32_16x16x4_f32(
                /*neg_a=*/false, av, /*neg_b=*/false, bv,
                /*c_mod=*/(short)0, c, /*reuse_a=*/false, /*reuse_b=*/false);
    }

    // C/D layout: VGPR r -> M=r (lanes 0-15) / M=r+8 (lanes 16-31), N=lane&15
    const int col   = ht * 16 + m;
    const int rbase = it * 16 + ((lane >> 4) ? 8 : 0);
    const float bias = ij ? b1[(size_t)k * Hh + col] : 0.0f;
    float* Out = (ij ? HJ : HI) + ((size_t)(k * Bb + b) * Nn) * Hh;
    #pragma unroll
    for (int r = 0; r < 8; ++r) {
        Out[(size_t)(rbase + r) * Hh + col] = c[r] + bias;
    }
}

// ---------------------------------------------------------------------------
// Kernel 2: pairwise relu + entmax15 + W3 projection.
// Block = 256 threads, 32x32 pair tile in LDS, 2x2 register blocking.
//   ldiff = sum_h relu(u_i[h] + v_j[h]) * (W2[h,0]-W2[h,1]) + (b2[0]-b2[1])
//   t = 0.25*ldiff ; p0 = entmax pair ; out = p0*(W3[0]-W3[1]) + W3[1] + b3
// ---------------------------------------------------------------------------
__global__ __launch_bounds__(256)
void fst_pair(const float* __restrict__ HI, const float* __restrict__ HJ,
              const float* __restrict__ W2, const float* __restrict__ b2,
              const float* __restrict__ W3, const float* __restrict__ b3,
              float* __restrict__ out)
{
    __shared__ float Us[32][132];   // pad to 132 floats: V reads hit distinct banks
    __shared__ float Vs[32][132];
    __shared__ float w2s[Hh];

    const int th = threadIdx.x;
    const int k  = blockIdx.y;
    const int b  = blockIdx.z;
    const int ti = blockIdx.x >> 3;
    const int tj = blockIdx.x & 7;
    const int i0 = ti * 32, j0 = tj * 32;

    const float* HIp = HI + (((size_t)k * Bb + b) * Nn + i0) * Hh;
    const float* HJp = HJ + (((size_t)k * Bb + b) * Nn + j0) * Hh;

    // Fill LDS: 32 rows x 128 cols each, float4 vectorized (4 per thread)
    #pragma unroll
    for (int t = 0; t < 4; ++t) {
        int idx = th + t * 256;        // 0..1023
        int row = idx >> 5;
        int c4  = (idx & 31) << 2;
        float4 u = *(const float4*)(HIp + (size_t)row * Hh + c4);
        float4 v = *(const float4*)(HJp + (size_t)row * Hh + c4);
        *(float4*)(&Us[row][c4]) = u;
        *(float4*)(&Vs[row][c4]) = v;
    }
    if (th < Hh) {
        w2s[th] = W2[((size_t)k * Hh + th) * 2 + 0] - W2[((size_t)k * Hh + th) * 2 + 1];
    }
    __syncthreads();

    // 2x2 register block: i in {il, il+1}, j in {jl, jl+16}
    const int jl = th & 15;
    const int il = (th >> 4) * 2;
    const float* u0p = Us[il];
    const float* u1p = Us[il + 1];
    const float* v0p = Vs[jl];
    const float* v1p = Vs[jl + 16];

    float a00 = 0.f, a01 = 0.f, a10 = 0.f, a11 = 0.f;
    #pragma unroll 8
    for (int h = 0; h < Hh; ++h) {
        float u0 = u0p[h], u1 = u1p[h];
        float v0 = v0p[h], v1 = v1p[h];
        float w  = w2s[h];
        a00 = fmaf(fmaxf(u0 + v0, 0.f), w, a00);
        a01 = fmaf(fmaxf(u0 + v1, 0.f), w, a01);
        a10 = fmaf(fmaxf(u1 + v0, 0.f), w, a10);
        a11 = fmaf(fmaxf(u1 + v1, 0.f), w, a11);
    }

    const float b2d = b2[k * 2 + 0] - b2[k * 2 + 1];
    const float w3d = W3[k * 2 + 0] - W3[k * 2 + 1];
    const float w3c = W3[k * 2 + 1] + b3[k];

    float accs[4] = {a00, a01, a10, a11};
    #pragma unroll
    for (int q = 0; q < 4; ++q) {
        int i = i0 + il + (q >> 1);
        int j = j0 + jl + ((q & 1) ? 16 : 0);
        float t  = 0.25f * (accs[q] + b2d);
        float dd = fabsf(t);
        float s  = sqrtf(fmaxf(0.5f - t * t, 0.f));
        float p0i = (s + t) * (s + t);
        float p0 = (dd >= 0.5f) ? ((t > 0.f) ? 1.f : 0.f) : p0i;
        float a  = fmaf(p0, w3d, w3c);
        if (i == j) a = 0.f;
        out[(((size_t)b * Kk + k) * Nn + i) * Nn + j] = a;
    }
}

// ---------------------------------------------------------------------------
extern "C" void kernel_launch(void* const* d_in, const int* in_sizes, int n_in,
                              void* d_out, int out_size, void* d_ws, size_t ws_size,
                              hipStream_t stream) {
    const float* E  = (const float*)d_in[0];
    const float* W1 = (const float*)d_in[1];
    const float* b1 = (const float*)d_in[2];
    const float* W2 = (const float*)d_in[3];
    const float* b2 = (const float*)d_in[4];
    const float* W3 = (const float*)d_in[5];
    const float* b3 = (const float*)d_in[6];
    float* out = (float*)d_out;

    float* HI = (float*)d_ws;                          // K*B*N*H floats (8 MB)
    float* HJ = HI + (size_t)Kk * Bb * Nn * Hh;        // another 8 MB

    // Stage 1: WMMA GEMMs (K heads x B batches x {I,J} x 16x16 tiles)
    dim3 g1(Kk * Bb * 2 * (Nn / 16) * (Hh / 16));      // 16384 waves
    fst_gemm_wmma<<<g1, 32, 0, stream>>>(E, W1, b1, HI, HJ);

    // Stage 2: pairwise relu/entmax/projection
    dim3 g2((Nn / 32) * (Nn / 32), Kk, Bb);            // 64 x 8 x 8 blocks
    fst_pair<<<g2, 256, 0, stream>>>(HI, HJ, W2, b2, W3, b3, out);
}